// LocalGirdAttention_52578989637705
// MI455X (gfx1250) — compile-verified
//
#include <hip/hip_runtime.h>
#include <math.h>

// ---------------------------------------------------------------------------
// CDNA5 fused local-grid attention (fp32-exact, V_WMMA_F32_16X16X4_F32).
//   Kernel 1: q = scaling*(Wq x + bq)   -> d_ws
//   Kernel 2: per 2x4 query tile: fused k-projection (WMMA), softmax,
//             attn-weighted featmap aggregate, then out = Wv*g + bv
//             (v-projection commuted past softmax: exact identity, saves
//              17 GFLOP + 536 MB of k/v HBM round trips; bk provably drops).
//   * pair-interleaved B-operand LDS layouts -> B frags are single
//     ds_load_b64, eliminating v_dual_mov repacking in the WMMA loop.
//   * per-head weight tiles staged by the Tensor Data Mover
//     (tensor_load_to_lds + s_wait_tensorcnt), double-buffered and
//     prefetched one phase ahead; manual-copy fallback if builtin absent.
//   * this toolchain exposes the 6-arg tensor_load_to_lds (clang-23 form):
//     (v4u g0, v8i g1, v4i g2, v4i g3, v8i g4, i32 cpol) -- g2/g3/g4 zero
//     for a 2D tile.
// ---------------------------------------------------------------------------

typedef float v2f __attribute__((ext_vector_type(2)));
typedef float v8f __attribute__((ext_vector_type(8)));

// V_WMMA_F32_16X16X4_F32: D[16x16] = A[16x4]*B[4x16] + C  (per-lane layouts in
// ISA 7.12.2: A lane l -> row l%16, K=2*(l/16)+{0,1} (contig float2);
// B lane l -> col l%16, rows 2*(l/16)+{0,1}; D vgpr g -> m=g+8*(l/16), n=l%16)
__device__ __forceinline__ v8f wmma4(v2f a, v2f b, v8f c) {
  return __builtin_amdgcn_wmma_f32_16x16x4_f32(
      false, a, false, b, (short)0, c, false, false);
}

#define Cc    256
#define HW    4096
#define NH    8
#define HD    32

#if defined(__has_builtin)
#if __has_builtin(__builtin_amdgcn_tensor_load_to_lds) && \
    __has_builtin(__builtin_amdgcn_s_wait_tensorcnt)
#define USE_TDM 1
#endif
#endif

#ifdef USE_TDM
typedef unsigned int v4u __attribute__((ext_vector_type(4)));
typedef int v8i __attribute__((ext_vector_type(8)));
typedef int v4i __attribute__((ext_vector_type(4)));

// TDM load of one weight head-slice: 2D tile 32 rows x 256 fp32, global row
// stride 256, LDS padded +4 dwords every 256 dwords -> 260-float row stride.
// D# per CDNA5 ISA 8.3-8.6 (count=1, type=2, data_size=4B, pad en).
__device__ __forceinline__ void tdm_load_w(const float* gsrc, unsigned lds_byte) {
  unsigned long long ga = (unsigned long long)(uintptr_t)gsrc;
  v4u g0;
  g0[0] = 1u;                                   // count=1 user descriptor
  g0[1] = lds_byte;                             // lds_addr
  g0[2] = (unsigned)(ga & 0xFFFFFFFFu);         // global_addr[31:0]
  g0[3] = (unsigned)((ga >> 32) & 0x1FFFFFFu) | (2u << 30);  // [56:32] | type=2
  v8i g1;
  g1[0] = (2 << 16) | (1 << 20) | (7 << 22) | (3 << 25);
  //        data=4B   pad_en      int=256dw    amount=4dw
  g1[1] = (int)(0x100u << 16);          // tensor_dim0 = 256 (lo16 at [63:48])
  g1[2] = (int)(0x100u << 16);          // td0 hi=0 | tensor_dim1 = 256 lo16
  g1[3] = (int)(0x100u << 16);          // td1 hi=0 | tile_dim0 = 256
  g1[4] = 32;                           // tile_dim1 = 32, tile_dim2 = 0 (2D)
  g1[5] = 256;                          // tensor_dim0_stride = 256
  g1[6] = 0;
  g1[7] = 0;
  v4i g2 = {0, 0, 0, 0};                // 2D tile: groups 2/3 unused
  v4i g3 = {0, 0, 0, 0};
  v8i g4 = {0, 0, 0, 0, 0, 0, 0, 0};
  __builtin_amdgcn_tensor_load_to_lds(g0, g1, g2, g3, g4, 0);
}
#endif

// ---------------- Kernel 1: Q projection --------------------------------
// sX pair-interleaved: (c,n) -> (c>>1)*K1XP + 2n + (c&1); 160 % 64 == 32 so
// the two lane-halves (pair-rows kk*2 and kk*2+1) hit disjoint bank sets.
#define K1N   64
#define K1XP  160
#define K1_LDS_FLOATS (128 * K1XP)

__global__ __launch_bounds__(256)
void lga_qproj(const float* __restrict__ x, const float* __restrict__ Wq,
               const float* __restrict__ bq, float* __restrict__ q) {
  extern __shared__ float sm[];
  const int tid  = threadIdx.x;
  const int lane = tid & 31, wave = tid >> 5;
  const int lm = lane & 15, lh = lane >> 4;
  const int b  = blockIdx.x >> 6;
  const int n0 = (blockIdx.x & 63) * K1N;

  {  // stage x tile [256 c][64 n], pair-interleaved
    const float* src = x + ((size_t)b * Cc + tid) * HW + n0;
    float* dst = &sm[(tid >> 1) * K1XP + (tid & 1)];
#pragma unroll
    for (int i = 0; i < 16; ++i) {
      float4 v = *(const float4*)(src + i * 4);
      dst[(i * 4 + 0) * 2] = v.x; dst[(i * 4 + 1) * 2] = v.y;
      dst[(i * 4 + 2) * 2] = v.z; dst[(i * 4 + 3) * 2] = v.w;
    }
  }
  __syncthreads();

  v8f acc[8] = {};
  for (int kk = 0; kk < 64; ++kk) {
    const int c0 = kk * 4 + 2 * lh;
    const int prow = kk * 2 + lh;
#pragma unroll
    for (int mh = 0; mh < 2; ++mh) {
      const int m = (wave * 2 + mh) * 16 + lm;
      v2f a = *(const v2f*)(Wq + m * Cc + c0);   // L2-resident weights
#pragma unroll
      for (int nt = 0; nt < 4; ++nt) {
        v2f bb = *(const v2f*)&sm[prow * K1XP + (nt * 16 + lm) * 2];
        acc[mh * 4 + nt] = wmma4(a, bb, acc[mh * 4 + nt]);
      }
    }
  }
  const float SCAL = 0.1767766952966369f;  // 32^-0.5
#pragma unroll
  for (int mh = 0; mh < 2; ++mh) {
    const int mt = wave * 2 + mh;
#pragma unroll
    for (int nt = 0; nt < 4; ++nt) {
#pragma unroll
      for (int g = 0; g < 8; ++g) {
        const int o = mt * 16 + g + 8 * lh;
        const int n = n0 + nt * 16 + lm;
        q[((size_t)b * Cc + o) * HW + n] = SCAL * (acc[mh * 4 + nt][g] + bq[o]);
      }
    }
  }
}

// ---------------- Kernel 2: fused local-grid attention -------------------
#define NQP   8
#define SFP   288   // sF pair-row stride (288%64==32: lane-halves disjoint)
#define WSs   260   // sW row stride (matches TDM pad: 256+4 dwords)
#define SKS   132   // sK row stride
#define SGP   32    // sG pair-row stride (32%64==32)
#define SAS   17
#define SOS   20
#define OFF_F  0
#define OFF_W0 (OFF_F + 128 * SFP)      // 36864
#define OFF_W1 (OFF_W0 + 32 * WSs)      // +8320
#define OFF_K  (OFF_W1 + 32 * WSs)      // +8320
#define OFF_Q  (OFF_K + 32 * SKS)       // +4224
#define OFF_G  (OFF_Q + 256 * 9)        // +2304
#define OFF_A  (OFF_G + 128 * SGP)      // +4096
#define OFF_O  (OFF_A + 8 * SAS)        // +136
#define LDS_FLOATS (OFF_O + 32 * SOS)   // 64904 floats = 259616 B (< 320 KB)

__device__ __forceinline__ void stage_w_manual(const float* W, int h,
                                               float* dst, int tid) {
  const int r = tid >> 3, cb = (tid & 7) * 32;
  const float* src = W + (h * HD + r) * Cc + cb;
#pragma unroll
  for (int i = 0; i < 8; ++i) {
    float4 v = *(const float4*)(src + i * 4);
    float* d = &dst[r * WSs + cb + i * 4];
    d[0] = v.x; d[1] = v.y; d[2] = v.z; d[3] = v.w;
  }
}

__global__ __launch_bounds__(256)
void lga_attn(const float* __restrict__ fm, const float* __restrict__ Wk,
              const float* __restrict__ Wv, const float* __restrict__ bv,
              const float* __restrict__ q, float* __restrict__ out) {
  extern __shared__ float sm[];
  const int tid  = threadIdx.x;
  const int lane = tid & 31, wave = tid >> 5;
  const int lm = lane & 15, lh = lane >> 4;
  const int bid = blockIdx.x;
  const int b   = bid >> 9;
  const int qby = (bid >> 4) & 31;
  const int qbx = bid & 15;
  const int y0 = qby * 8, x0 = qbx * 16;

#ifdef USE_TDM
  const unsigned lds_base = __builtin_amdgcn_groupstaticsize();
  if (wave == 0)  // prefetch Wk_0 while the featmap tile streams in
    tdm_load_w(Wk, lds_base + OFF_W0 * 4u);
#endif

  // ---- stage featmap tile [256 c][8 y][16 x], pair-interleaved over c
#pragma unroll
  for (int i = 0; i < 32; ++i) {
    const int idx = tid + 256 * i;
    const int c = idx >> 5, r = idx & 31;
    const int ty = r >> 2, txv = r & 3;
    float4 v = *(const float4*)(fm + (((size_t)b * Cc + c) * 256 + y0 + ty) * 256 + x0 + txv * 4);
    float* d = &sm[OFF_F + (c >> 1) * SFP + (ty * 16 + txv * 4) * 2 + (c & 1)];
    d[0] = v.x; d[2] = v.y; d[4] = v.z; d[6] = v.w;
  }
  // ---- stage q block: sQ[d][qp] (stride 9), qp = qy*4+qx
#pragma unroll
  for (int i = 0; i < 2; ++i) {
    const int idx = tid + 256 * i;
    const int d = idx >> 1, qy = idx & 1;
    float4 v = *(const float4*)(q + (((size_t)b * Cc + d) * 64 + qby * 2 + qy) * 64 + qbx * 4);
    float* dd = &sm[OFF_Q + d * 9 + qy * 4];
    dd[0] = v.x; dd[1] = v.y; dd[2] = v.z; dd[3] = v.w;
  }
  __syncthreads();

  for (int h = 0; h < NH; ++h) {
#ifdef USE_TDM
    if (wave == 0) {
      tdm_load_w(Wv + (size_t)h * HD * Cc, lds_base + OFF_W1 * 4u);
      __builtin_amdgcn_s_wait_tensorcnt(1);   // in-order: Wk_h retired
    }
#else
    stage_w_manual(Wk, h, &sm[OFF_W0], tid);
#endif
    __syncthreads();
    // ---- k-projection: Kh[32][128] = Wk_h * F (16 WMMA jobs, 8 waves)
    {
      const int mt  = wave >> 2;
      const int nt0 = (wave & 3) * 2, nt1 = nt0 + 1;
      v8f acc0 = {}, acc1 = {};
      for (int kk = 0; kk < 64; ++kk) {
        const int prow = kk * 2 + lh;
        v2f a = *(const v2f*)&sm[OFF_W0 + (mt * 16 + lm) * WSs + kk * 4 + 2 * lh];
        v2f b0 = *(const v2f*)&sm[OFF_F + prow * SFP + (nt0 * 16 + lm) * 2];
        acc0 = wmma4(a, b0, acc0);
        v2f b1 = *(const v2f*)&sm[OFF_F + prow * SFP + (nt1 * 16 + lm) * 2];
        acc1 = wmma4(a, b1, acc1);
      }
#pragma unroll
      for (int g = 0; g < 8; ++g) {
        const int m = mt * 16 + g + 8 * lh;
        sm[OFF_K + m * SKS + nt0 * 16 + lm] = acc0[g];
        sm[OFF_K + m * SKS + nt1 * 16 + lm] = acc1[g];
      }
    }
    __syncthreads();
#ifdef USE_TDM
    if (wave == 0 && h < NH - 1)   // prefetch next head's Wk during VALU phases
      tdm_load_w(Wk + (size_t)(h + 1) * HD * Cc, lds_base + OFF_W0 * 4u);
#endif
    // ---- logits: thread (qp,p) does a 32-dot (bk is softmax-invariant)
    if (tid < 128) {
      const int qp = tid >> 4, p = tid & 15;
      const int n = ((qp >> 2) * 4 + (p >> 2)) * 16 + (qp & 3) * 4 + (p & 3);
      float s = 0.f;
#pragma unroll
      for (int d = 0; d < HD; ++d)
        s += sm[OFF_Q + (h * HD + d) * 9 + qp] * sm[OFF_K + d * SKS + n];
      sm[OFF_A + qp * SAS + p] = s;
    }
    __syncthreads();
    // ---- softmax over 16 local positions (one thread per query pixel)
    if (tid < NQP) {
      float mx = -1e30f;
#pragma unroll
      for (int p = 0; p < 16; ++p) mx = fmaxf(mx, sm[OFF_A + tid * SAS + p]);
      float e[16], ssum = 0.f;
#pragma unroll
      for (int p = 0; p < 16; ++p) {
        e[p] = __expf(sm[OFF_A + tid * SAS + p] - mx);
        ssum += e[p];
      }
      const float inv = 1.f / ssum;
#pragma unroll
      for (int p = 0; p < 16; ++p) sm[OFF_A + tid * SAS + p] = e[p] * inv;
    }
    __syncthreads();
    // ---- attn-weighted featmap aggregate g[c][qp] (+Wv stage on fallback)
    {
#ifndef USE_TDM
      stage_w_manual(Wv, h, &sm[OFF_W1], tid);
#endif
      const int c = tid;
      const int fbase = OFF_F + (c >> 1) * SFP + (c & 1);
      float gg[NQP];
#pragma unroll
      for (int qp = 0; qp < NQP; ++qp) gg[qp] = 0.f;
#pragma unroll
      for (int p = 0; p < 16; ++p) {
#pragma unroll
        for (int qp = 0; qp < NQP; ++qp) {
          const int n = ((qp >> 2) * 4 + (p >> 2)) * 16 + (qp & 3) * 4 + (p & 3);
          gg[qp] += sm[OFF_A + qp * SAS + p] * sm[fbase + n * 2];
        }
      }
      const int gbase = OFF_G + (c >> 1) * SGP + (c & 1);
#pragma unroll
      for (int qp = 0; qp < NQP; ++qp) sm[gbase + qp * 2] = gg[qp];
#pragma unroll
      for (int qp = NQP; qp < 16; ++qp) sm[gbase + qp * 2] = 0.f;
      for (int i = tid; i < 32 * SOS; i += 256) sm[OFF_O + i] = 0.f;
    }
#ifdef USE_TDM
    if (wave == 0) {
      if (h < NH - 1) __builtin_amdgcn_s_wait_tensorcnt(1);  // Wv_h retired
      else            __builtin_amdgcn_s_wait_tensorcnt(0);
    }
#endif
    __syncthreads();
    // ---- out GEMM: [32x16] = Wv_h[32x256] * g[256x16], 4-way split-K,
    //      ds_add_f32 reduction into sOut
    {
      const int mt = wave & 1;
      const int ks = (wave >> 1) * 16;
      v8f acc = {};
      for (int kk = ks; kk < ks + 16; ++kk) {
        v2f a = *(const v2f*)&sm[OFF_W1 + (mt * 16 + lm) * WSs + kk * 4 + 2 * lh];
        v2f bb = *(const v2f*)&sm[OFF_G + (kk * 2 + lh) * SGP + lm * 2];
        acc = wmma4(a, bb, acc);
      }
#pragma unroll
      for (int g = 0; g < 8; ++g) {
        const int m = mt * 16 + g + 8 * lh;
        atomicAdd(&sm[OFF_O + m * SOS + lm], acc[g]);
      }
    }
    __syncthreads();
    // ---- epilogue: out = sOut + bv (softmax weights sum to 1)
    {
      const int m = tid >> 3, qp = tid & 7;
      const int och = h * HD + m;
      const float val = sm[OFF_O + m * SOS + qp] + bv[och];
      const int oy = qby * 2 + (qp >> 2);
      const int ox = qbx * 4 + (qp & 3);
      out[(((size_t)b * Cc + och) * 64 + oy) * 64 + ox] = val;
    }
    __syncthreads();
  }
}

// ---------------------------------------------------------------------------
extern "C" void kernel_launch(void* const* d_in, const int* in_sizes, int n_in,
                              void* d_out, int out_size, void* d_ws, size_t ws_size,
                              hipStream_t stream) {
  const float* x  = (const float*)d_in[0];
  const float* fm = (const float*)d_in[1];
  const float* Wq = (const float*)d_in[2];
  const float* bq = (const float*)d_in[3];
  const float* Wk = (const float*)d_in[4];
  // d_in[5] = bk: uniform logit shift, softmax-invariant -> unused
  const float* Wv = (const float*)d_in[6];
  const float* bv = (const float*)d_in[7];
  float* out  = (float*)d_out;
  float* qbuf = (float*)d_ws;  // 8 MB scratch for q
  (void)in_sizes; (void)n_in; (void)out_size; (void)ws_size;

  lga_qproj<<<dim3(2 * 64), dim3(256), K1_LDS_FLOATS * sizeof(float), stream>>>(
      x, Wq, bq, qbuf);
  lga_attn<<<dim3(2 * 32 * 16), dim3(256), LDS_FLOATS * sizeof(float), stream>>>(
      fm, Wk, Wv, bv, qbuf, out);
}